// PopulationGCN_55259049230385
// MI455X (gfx1250) — compile-verified
//
#include <hip/hip_runtime.h>

// GCN: h1 = relu(gcnconv(x,W1,b1)); h2 = relu(gcnconv(h1,W2,b2)); out = h2@Wfc + bfc
// gcnconv: h = X@W ; deg = scatter(ew)+1 ; dinv = rsqrt(deg) ;
//          agg = scatter_add(h[src]*dinv[src]*ew*dinv[dst] -> dst) + h*dinv^2 + b

typedef __attribute__((ext_vector_type(2))) float v2f;
typedef __attribute__((ext_vector_type(8))) float v8f;

#define F_IN 128
#define F_H  64
#define F_OUT 10

// ---------------- utility kernels ----------------

__global__ void gcn_zero_f32(float* __restrict__ p, long long n) {
  long long i = (long long)blockIdx.x * blockDim.x + threadIdx.x;
  if (i < n) p[i] = 0.0f;
}

__global__ void gcn_deg_accum(const long long* __restrict__ dst,
                              const float* __restrict__ ew,
                              float* __restrict__ deg, long long E) {
  long long e = (long long)blockIdx.x * blockDim.x + threadIdx.x;
  if (e < E) atomicAdd(&deg[dst[e]], ew[e]);
}

__global__ void gcn_make_dinv(float* __restrict__ deg, int n) {
  int i = blockIdx.x * blockDim.x + threadIdx.x;
  if (i < n) deg[i] = rsqrtf(deg[i] + 1.0f);   // self-loop weight 1 included
}

__global__ void gcn_make_norm(const long long* __restrict__ src,
                              const long long* __restrict__ dst,
                              const float* __restrict__ ew,
                              const float* __restrict__ dinv,
                              float* __restrict__ norm, long long E) {
  long long e = (long long)blockIdx.x * blockDim.x + threadIdx.x;
  if (e < E) norm[e] = dinv[src[e]] * ew[e] * dinv[dst[e]];
}

// ---------------- WMMA fp32 GEMM: C[M,NOUT] = A[M,K] @ W[K,NOUT] ----------------
// One wave32 computes one 16x16 output tile with V_WMMA_F32_16X16X4_F32.
// ISA VGPR layouts (CDNA5 ISA):
//   A 16x4 : lanes 0-15 -> M=lane, v0=K(2g), v1=K(2g+1) where g = lane>>4
//   B 4x16 : v0 holds rows K=2g across lanes (N=lane&15), v1 holds K=2g+1
//   C/D    : vgpr i -> M = 16*mt + i + 8*g, N = 16*nt + (lane&15)
// Requires M % 16 == 0 (true here: 100000), NOUT % 16 == 0, K % 4 == 0.
template <int K, int NOUT>
__global__ void gcn_gemm_wmma_f32(const float* __restrict__ A,
                                  const float* __restrict__ W,
                                  float* __restrict__ C, int M) {
  const int lane = threadIdx.x & 31;
  const int wave = (int)((blockIdx.x * blockDim.x + threadIdx.x) >> 5);
  const int ntiles = NOUT / 16;
  const int mt = wave / ntiles;
  const int nt = wave - mt * ntiles;
  if (mt * 16 >= M) return;                 // wave-uniform: EXEC stays all-1s
  const int g = lane >> 4;                  // half-wave group (0/1)
  const int l = lane & 15;
  const long long row = (long long)mt * 16 + l;
  const int col = nt * 16 + l;

  v8f acc = {};
#pragma unroll
  for (int kk = 0; kk < K; kk += 4) {
    const int ka = kk + 2 * g;
    v2f a, b;
    a.x = A[row * K + ka];
    a.y = A[row * K + ka + 1];
    b.x = W[ka * NOUT + col];
    b.y = W[(ka + 1) * NOUT + col];
    // (neg_a, A, neg_b, B, c_mod, C, reuse_a, reuse_b)
    acc = __builtin_amdgcn_wmma_f32_16x16x4_f32(false, a, false, b,
                                                (short)0, acc, false, false);
  }

  const long long rbase = (long long)mt * 16 + 8 * g;
#pragma unroll
  for (int i = 0; i < 8; ++i)
    C[(rbase + i) * NOUT + col] = acc[i];
}

// ---------------- edge scatter-add (one thread per edge-feature) ----------------
__global__ void gcn_scatter_agg(const float* __restrict__ H,
                                const long long* __restrict__ src,
                                const long long* __restrict__ dst,
                                const float* __restrict__ norm,
                                float* __restrict__ agg, long long E) {
  long long i = (long long)blockIdx.x * blockDim.x + threadIdx.x;
  if (i >= (E << 6)) return;                // 64 features per edge
  long long e = i >> 6;
  int f = (int)(i & 63);
  long long s = src[e], d = dst[e];
  atomicAdd(&agg[d * F_H + f], H[s * F_H + f] * norm[e]);
}

// ---------------- self-loop + bias + relu, in place on agg ----------------
__global__ void gcn_combine_relu(float* __restrict__ agg,
                                 const float* __restrict__ H,
                                 const float* __restrict__ dinv,
                                 const float* __restrict__ bias, long long n64) {
  long long i = (long long)blockIdx.x * blockDim.x + threadIdx.x;
  if (i >= n64) return;
  int r = (int)(i >> 6);
  int f = (int)(i & 63);
  float di = dinv[r];
  float v = agg[i] + H[i] * (di * di) + bias[f];
  agg[i] = v > 0.0f ? v : 0.0f;
}

// ---------------- final FC head (N=10, too ragged for WMMA; tiny FLOPs) ----------------
__global__ void gcn_fc_out(const float* __restrict__ H, const float* __restrict__ W,
                           const float* __restrict__ b, float* __restrict__ out, int M) {
  int i = blockIdx.x * blockDim.x + threadIdx.x;
  if (i >= M * F_OUT) return;
  int r = i / F_OUT;
  int c = i - r * F_OUT;
  float s = b[c];
#pragma unroll
  for (int k = 0; k < F_H; ++k)
    s = fmaf(H[(long long)r * F_H + k], W[k * F_OUT + c], s);
  out[i] = s;
}

// ---------------- launcher ----------------
extern "C" void kernel_launch(void* const* d_in, const int* in_sizes, int n_in,
                              void* d_out, int out_size, void* d_ws, size_t ws_size,
                              hipStream_t stream) {
  const float*     x   = (const float*)d_in[0];
  const long long* ei  = (const long long*)d_in[1];   // int64 [2, E]
  const float*     ew  = (const float*)d_in[2];
  const float*     W1  = (const float*)d_in[3];
  const float*     b1  = (const float*)d_in[4];
  const float*     W2  = (const float*)d_in[5];
  const float*     b2  = (const float*)d_in[6];
  const float*     Wfc = (const float*)d_in[7];
  const float*     bfc = (const float*)d_in[8];
  float*           out = (float*)d_out;

  const int       N = in_sizes[0] / F_IN;    // 100000
  const long long E = in_sizes[2];           // 1600000
  const long long* src = ei;
  const long long* dst = ei + E;

  // workspace layout: hbuf[N*64] | abuf[N*64] | norm[E] | dinv[N]  (~58 MB)
  float* hbuf = (float*)d_ws;
  float* abuf = hbuf + (size_t)N * F_H;
  float* norm = abuf + (size_t)N * F_H;
  float* dinv = norm + (size_t)E;

  const int T = 256;
  auto nb = [](long long n, int t) { return (unsigned)((n + t - 1) / t); };

  const long long nh   = (long long)N * F_H;
  const long long nef  = E << 6;
  const int mtiles     = (N + 15) / 16;
  const long long gemmThreads = (long long)mtiles * (F_H / 16) * 32;

  // ---- shared normalization ----
  gcn_zero_f32   <<<nb(N, T), T, 0, stream>>>(dinv, N);
  gcn_deg_accum  <<<nb(E, T), T, 0, stream>>>(dst, ew, dinv, E);
  gcn_make_dinv  <<<nb(N, T), T, 0, stream>>>(dinv, N);
  gcn_make_norm  <<<nb(E, T), T, 0, stream>>>(src, dst, ew, dinv, norm, E);

  // ---- layer 1 ----
  gcn_gemm_wmma_f32<F_IN, F_H><<<nb(gemmThreads, T), T, 0, stream>>>(x, W1, hbuf, N);
  gcn_zero_f32    <<<nb(nh, T), T, 0, stream>>>(abuf, nh);
  gcn_scatter_agg <<<nb(nef, T), T, 0, stream>>>(hbuf, src, dst, norm, abuf, E);
  gcn_combine_relu<<<nb(nh, T), T, 0, stream>>>(abuf, hbuf, dinv, b1, nh);

  // ---- layer 2 (abuf = h1 post-relu; hbuf reused for h2 pre-agg) ----
  gcn_gemm_wmma_f32<F_H, F_H><<<nb(gemmThreads, T), T, 0, stream>>>(abuf, W2, hbuf, N);
  gcn_zero_f32    <<<nb(nh, T), T, 0, stream>>>(abuf, nh);   // stream order: after GEMM2 reads abuf
  gcn_scatter_agg <<<nb(nef, T), T, 0, stream>>>(hbuf, src, dst, norm, abuf, E);
  gcn_combine_relu<<<nb(nh, T), T, 0, stream>>>(abuf, hbuf, dinv, b2, nh);

  // ---- FC head ----
  gcn_fc_out<<<nb((long long)N * F_OUT, T), T, 0, stream>>>(abuf, Wfc, bfc, out, N);
}